// GradientLoss_56478819942625
// MI455X (gfx1250) — compile-verified
//
#include <hip/hip_runtime.h>
#include <stdint.h>

// ---------------------------------------------------------------------------
// GradientLoss (3D Sobel MSE) for MI455X / gfx1250.
// Bandwidth-bound: 110 MB read @ 23.3 TB/s ~= 4.7 us floor; fits in 192MB L2.
// Uses CDNA5 async global->LDS (GLOBAL_LOAD_ASYNC_TO_LDS_B128 + s_wait_asynccnt)
// to stage halo tiles, then separable stencil from LDS with sliding windows.
// ---------------------------------------------------------------------------

#define N_DIM 2
#define D_DIM 160
#define H_DIM 192
#define W_DIM 224

#define TD 4                 // tile depth  (outputs)
#define THH 8                // tile height (outputs)
#define HD (TD + 2)          // halo depth   = 6
#define HHH (THH + 2)        // halo height  = 10
#define NROWS (HD * HHH)     // 60 rows per input tile
#define ROWSTRIDE 232        // floats per LDS row: [0..3] pad, [3]=0-guard(w=-1),
                             // [4..227]=w 0..223, [228]=0-guard(w=224); 928B = 16B-aligned
#define TILE_FLOATS (NROWS * ROWSTRIDE)   // 13920 floats = 55680 B per input

#define NTHREADS 256
#define STRIPS 8
#define STRIPLEN (W_DIM / STRIPS)         // 28

__device__ __forceinline__ void async_load_b128(uint32_t lds_byte_off, const float* gptr) {
  // CDNA5 async copy: memory -> LDS, tracked by ASYNCcnt (no VGPR staging).
  asm volatile("global_load_async_to_lds_b128 %0, %1, off"
               :
               : "v"(lds_byte_off), "v"((unsigned long long)(uintptr_t)gptr)
               : "memory");
}

__device__ __forceinline__ void wait_async_all() {
  asm volatile("s_wait_asynccnt 0x0" ::: "memory");
}

__device__ __forceinline__ uint32_t lds_off_of(const void* p) {
  // Low 32 bits of a flat pointer in the LDS aperture == LDS byte address.
  return (uint32_t)(uintptr_t)p;
}

// Column functional: from 9 values of v in the 3x3 (d,h) cross-section at one
// W column, produce P = S_h(v)[d-1]-S_h(v)[d+1], Q = S_d(D_h(v)), R = S_d(S_h(v)).
// gx = S_w(P), gy = S_w(Q), gz = D_w(R).
__device__ __forceinline__ void col_pqr(const float* __restrict__ c,
                                        float& P, float& Q, float& R) {
  float v00 = c[0];
  float v01 = c[ROWSTRIDE];
  float v02 = c[2 * ROWSTRIDE];
  float v10 = c[HHH * ROWSTRIDE];
  float v11 = c[(HHH + 1) * ROWSTRIDE];
  float v12 = c[(HHH + 2) * ROWSTRIDE];
  float v20 = c[2 * HHH * ROWSTRIDE];
  float v21 = c[(2 * HHH + 1) * ROWSTRIDE];
  float v22 = c[(2 * HHH + 2) * ROWSTRIDE];
  float cs0 = (v00 + v02) + 2.0f * v01;   // S_h at d-1
  float cs1 = (v10 + v12) + 2.0f * v11;   // S_h at d
  float cs2 = (v20 + v22) + 2.0f * v21;   // S_h at d+1
  float cd0 = v00 - v02;                  // D_h at d-1
  float cd1 = v10 - v12;
  float cd2 = v20 - v22;
  P = cs0 - cs2;
  Q = (cd0 + cd2) + 2.0f * cd1;
  R = (cs0 + cs2) + 2.0f * cs1;
}

__global__ __launch_bounds__(NTHREADS)
void sobel3d_mse_kernel(const float* __restrict__ x, const float* __restrict__ y,
                        float* __restrict__ block_partials) {
  extern __shared__ float smem[];
  float* xs = smem;                 // later overwritten with v = x - y
  float* ys = smem + TILE_FLOATS;

  const int tid = threadIdx.x;
  const int h0 = blockIdx.x * THH;
  const int d0 = blockIdx.y * TD;
  const int n  = blockIdx.z;

  // ---------------- Phase A: async-stage x,y halo tiles into LDS ----------
  const int Q4 = W_DIM / 4;         // 56 float4 chunks per row
  const uint32_t xs_off = lds_off_of(xs);
  const uint32_t ys_off = lds_off_of(ys);
  for (int t = tid; t < 2 * NROWS * Q4; t += NTHREADS) {
    int inp = t / (NROWS * Q4);
    int rr  = t - inp * (NROWS * Q4);
    int row = rr / Q4;
    int c4  = rr - row * Q4;
    int dd  = row / HHH;
    int hh  = row - dd * HHH;
    int gd  = d0 - 1 + dd;
    int gh  = h0 - 1 + hh;
    if ((unsigned)gd < (unsigned)D_DIM && (unsigned)gh < (unsigned)H_DIM) {
      const float* src = (inp ? y : x)
          + (((size_t)n * D_DIM + gd) * H_DIM + gh) * (size_t)W_DIM + (size_t)(c4 * 4);
      uint32_t dst = (inp ? ys_off : xs_off)
          + (uint32_t)(row * ROWSTRIDE + 4 + c4 * 4) * 4u;   // 16B-aligned
      async_load_b128(dst, src);
    }
  }
  wait_async_all();      // this wave's async copies landed in LDS
  __syncthreads();       // all waves' copies visible

  // ---------------- Phase B: v = x - y (0 outside volume), in place -------
  for (int e = tid; e < NROWS * W_DIM; e += NTHREADS) {
    int row = e / W_DIM;
    int col = e - row * W_DIM;
    int dd  = row / HHH;
    int hh  = row - dd * HHH;
    int gd  = d0 - 1 + dd;
    int gh  = h0 - 1 + hh;
    bool ok = (unsigned)gd < (unsigned)D_DIM && (unsigned)gh < (unsigned)H_DIM;
    int idx = row * ROWSTRIDE + 4 + col;
    float v = xs[idx] - ys[idx];
    xs[idx] = ok ? v : 0.0f;       // SAME zero padding for D/H halo rows
  }
  for (int r = tid; r < NROWS; r += NTHREADS) {
    xs[r * ROWSTRIDE + 3] = 0.0f;            // w = -1 guard
    xs[r * ROWSTRIDE + 4 + W_DIM] = 0.0f;    // w = 224 guard
  }
  __syncthreads();

  // ---------------- Phase C: separable stencil, sliding along W -----------
  const int d_l   = tid >> 6;          // [0,4)
  const int h_l   = (tid >> 3) & 7;    // [0,8)
  const int strip = tid & 7;           // [0,8)
  const float* vb = xs + (d_l * HHH + h_l) * ROWSTRIDE + 4 + strip * STRIPLEN;

  float pm, qm, rm, pc, qc, rc;
  col_pqr(vb - 1, pm, qm, rm);
  col_pqr(vb,     pc, qc, rc);

  float acc = 0.0f;
  #pragma unroll 4
  for (int w = 0; w < STRIPLEN; ++w) {
    float pp, qp, rp;
    col_pqr(vb + w + 1, pp, qp, rp);
    float gx = (pm + pp) + 2.0f * pc;    // S_w(P) = D_d(S_h(S_w v))
    float gy = (qm + qp) + 2.0f * qc;    // S_w(Q) = S_d(D_h(S_w v))
    float gz = rm - rp;                  // D_w(R) = S_d(S_h(D_w v))
    acc = fmaf(gx, gx, acc);
    acc = fmaf(gy, gy, acc);
    acc = fmaf(gz, gz, acc);
    pm = pc; pc = pp;
    qm = qc; qc = qp;
    rm = rc; rc = rp;
  }

  // ---------------- Block reduction (wave32 shuffles + LDS) ---------------
  for (int o = 16; o > 0; o >>= 1) acc += __shfl_down(acc, o, 32);
  __shared__ float red[NTHREADS / 32];
  if ((tid & 31) == 0) red[tid >> 5] = acc;
  __syncthreads();
  if (tid == 0) {
    float s = 0.0f;
    #pragma unroll
    for (int i = 0; i < NTHREADS / 32; ++i) s += red[i];
    block_partials[(blockIdx.z * gridDim.y + blockIdx.y) * gridDim.x + blockIdx.x] = s;
  }
}

__global__ __launch_bounds__(NTHREADS)
void final_reduce_kernel(const float* __restrict__ partials, int n, float scale,
                         float* __restrict__ out) {
  float s = 0.0f;
  for (int i = threadIdx.x; i < n; i += NTHREADS) s += partials[i];
  for (int o = 16; o > 0; o >>= 1) s += __shfl_down(s, o, 32);
  __shared__ float red[NTHREADS / 32];
  if ((threadIdx.x & 31) == 0) red[threadIdx.x >> 5] = s;
  __syncthreads();
  if (threadIdx.x == 0) {
    float t = 0.0f;
    #pragma unroll
    for (int i = 0; i < NTHREADS / 32; ++i) t += red[i];
    out[0] = t * scale;
  }
}

extern "C" void kernel_launch(void* const* d_in, const int* in_sizes, int n_in,
                              void* d_out, int out_size, void* d_ws, size_t ws_size,
                              hipStream_t stream) {
  const float* x = (const float*)d_in[0];
  const float* y = (const float*)d_in[1];
  float* partials = (float*)d_ws;              // 1920 floats of scratch

  dim3 grid(H_DIM / THH, D_DIM / TD, N_DIM);   // 24 x 40 x 2 = 1920 blocks
  const size_t shbytes = (size_t)2 * TILE_FLOATS * sizeof(float);  // 111,360 B

  sobel3d_mse_kernel<<<grid, NTHREADS, shbytes, stream>>>(x, y, partials);

  const int nblocks = (int)(grid.x * grid.y * grid.z);
  const float scale =
      1.0f / (float)((size_t)N_DIM * D_DIM * H_DIM * W_DIM);  // mean over N,D,H,W per channel
  final_reduce_kernel<<<1, NTHREADS, 0, stream>>>(partials, nblocks, scale, (float*)d_out);
}